// GCN_57509612093613
// MI455X (gfx1250) — compile-verified
//
#include <hip/hip_runtime.h>
#include <math.h>

// GCN layer, fused: scale = rsqrt(colsum(self_loop(mask))); out = relu(diag(scale) @ x @ W + b)
// f32 end-to-end using CDNA5 V_WMMA_F32_16X16X4_F32.

typedef float v2f __attribute__((ext_vector_type(2)));
typedef float v8f __attribute__((ext_vector_type(8)));

namespace {
constexpr int NN = 128;      // nodes: H == W == in_dim
constexpr int OD = 256;      // out_dim
constexpr int KT = 32;       // k-tiles of depth 4 (128 / 4)
constexpr int CTH = 8;       // column tiles per LDS staging pass (8 * 16 = 128 cols)
}

__global__ __launch_bounds__(256)
void gcn_fused_wmma(const float* __restrict__ x,
                    const float* __restrict__ mask,
                    const float* __restrict__ w,
                    const float* __restrict__ bias,
                    float* __restrict__ out) {
  // Weight fragments for 8 column-tiles: KT * CTH * 32 lanes * 2 floats = 64 KB
  __shared__ float lds_w[KT * CTH * 32 * 2];
  __shared__ float lds_part[256];
  __shared__ float lds_scale[NN];

  const int n    = blockIdx.x;
  const int tid  = threadIdx.x;
  const int lane = tid & 31;
  const int wid  = tid >> 5;           // 8 waves
  const int lhi  = (lane >> 4) & 1;    // upper half-wave?
  const int ln   = lane & 15;
  const int baseM = wid * 16;          // 8 waves * 16 rows = 128 rows

  const float* xm = x    + (size_t)n * NN * NN;
  const float* mk = mask + (size_t)n * NN * NN;
  float*       on = out  + (size_t)n * NN * OD;

  // ---- Phase A: partial column sums of self-loop mask (d = sum over rows h) ----
  {
    const int r    = tid & 127;
    const int half = tid >> 7;
    const int h0   = half * 64;
    float s = 0.0f;
    for (int h = h0; h < h0 + 64; ++h) {
      float v = mk[h * NN + r];
      if (h == r) v = 1.0f;            // self-loop value
      s += v;
    }
    lds_part[tid] = s;
  }

  // ---- Phase B: stage weight fragments for column tiles [0, 8) ----
  // Fragment order: per (kt, ct, lane) store {w[k0][col], w[k0+1][col]},
  // k0 = 4*kt + (lane>=16 ? 2 : 0), col = ct*16 + (lane&15)  (B-matrix 4x16 f32 layout).
  for (int idx = tid; idx < KT * CTH * 32; idx += 256) {
    const int l   = idx & 31;
    const int ctl = (idx >> 5) & (CTH - 1);
    const int kt  = idx >> 8;
    const int k0  = 4 * kt + ((l & 16) ? 2 : 0);
    const int col = ctl * 16 + (l & 15);
    v2f frag;
    frag.x = w[(k0    ) * OD + col];
    frag.y = w[(k0 + 1) * OD + col];
    *(v2f*)&lds_w[idx * 2] = frag;
  }

  // ---- Load A fragments for this wave's 16-row strip (x read exactly once) ----
  // Lane l<16: row baseM+l, K pair (4kt, 4kt+1); lanes 16-31: K pair (4kt+2, 4kt+3).
  v2f afrag[KT];
  {
    const int row = baseM + ln;
    const float* xr = xm + row * NN + lhi * 2;
    __builtin_prefetch(xr, 0, 0);
#pragma unroll
    for (int kt = 0; kt < KT; ++kt)
      afrag[kt] = *(const v2f*)(xr + kt * 4);
  }

  __syncthreads();

  // ---- Finalize scale = rsqrt(d) ----
  if (tid < 128)
    lds_scale[tid] = rsqrtf(lds_part[tid] + lds_part[tid + 128]);

  __syncthreads();

  // Per-lane row scales for accumulator rows (C/D layout: VGPR v -> row v + 8*lhi)
  float srow[8];
#pragma unroll
  for (int v = 0; v < 8; ++v)
    srow[v] = lds_scale[baseM + v + lhi * 8];

  // ---- GEMM: two staging passes of 8 column tiles each ----
#pragma unroll 1
  for (int half = 0; half < 2; ++half) {
    if (half == 1) {
      __syncthreads();                 // everyone done reading pass-0 fragments
      for (int idx = tid; idx < KT * CTH * 32; idx += 256) {
        const int l   = idx & 31;
        const int ctl = (idx >> 5) & (CTH - 1);
        const int kt  = idx >> 8;
        const int k0  = 4 * kt + ((l & 16) ? 2 : 0);
        const int col = (CTH + ctl) * 16 + (l & 15);
        v2f frag;
        frag.x = w[(k0    ) * OD + col];
        frag.y = w[(k0 + 1) * OD + col];
        *(v2f*)&lds_w[idx * 2] = frag;
      }
      __syncthreads();
    }

#pragma unroll 1
    for (int ctl = 0; ctl < CTH; ++ctl) {
      const int ct = half * CTH + ctl;
      v8f acc = {};
#pragma unroll
      for (int kt = 0; kt < KT; ++kt) {
        v2f b = *(const v2f*)&lds_w[((kt * CTH + ctl) * 32 + lane) * 2];
        acc = __builtin_amdgcn_wmma_f32_16x16x4_f32(
            /*neg_a=*/false, afrag[kt], /*neg_b=*/false, b,
            /*c_mod=*/(short)0, acc, /*reuse_a=*/false, /*reuse_b=*/false);
      }
      // Epilogue: out = relu(scale[row] * acc + bias[col])
      const int colg = ct * 16 + ln;
      const float bcol = bias[colg];
#pragma unroll
      for (int v = 0; v < 8; ++v) {
        float val = srow[v] * acc[v] + bcol;
        val = val > 0.0f ? val : 0.0f;
        on[(baseM + v + lhi * 8) * OD + colg] = val;
      }
    }
  }
}

extern "C" void kernel_launch(void* const* d_in, const int* in_sizes, int n_in,
                              void* d_out, int out_size, void* d_ws, size_t ws_size,
                              hipStream_t stream) {
  const float* x    = (const float*)d_in[0];  // [B,L,128,128]
  const float* mask = (const float*)d_in[1];  // [B,L,128,128]
  const float* w    = (const float*)d_in[2];  // [128,256]
  const float* bias = (const float*)d_in[3];  // [256]
  float* out = (float*)d_out;                 // [B,L,128,256]

  const int nbatch = in_sizes[0] / (NN * NN); // B*L = 1024
  gcn_fused_wmma<<<nbatch, 256, 0, stream>>>(x, mask, w, bias, out);
}